// GAT_And_GCN_layer_53807350284444
// MI455X (gfx1250) — compile-verified
//
#include <hip/hip_runtime.h>
#include <hip/hip_bf16.h>

// ---------------------------------------------------------------------------
// GNN forward (GAT + GCN + GraphConv + pooling + MLP head) for gfx1250.
// Dense matmuls use V_WMMA_F32_16X16X4_F32 (exact f32 on the matrix pipe).
// Edge aggregation is atomic-based; scatter targets (51MB) live in the 192MB L2.
// ---------------------------------------------------------------------------

typedef float v2f __attribute__((ext_vector_type(2)));
typedef float v8f __attribute__((ext_vector_type(8)));

#define HD 8     // heads
#define DD 128   // hidden dim

static __host__ __device__ inline int cdiv_i(int a, int b) { return (a + b - 1) / b; }

// ---- monotone float<->uint encoding for atomicMax on floats -----------------
__device__ inline unsigned enc_f(float f) {
  unsigned u = __float_as_uint(f);
  return (u & 0x80000000u) ? ~u : (u | 0x80000000u);
}
__device__ inline float dec_f(unsigned u) {
  return (u & 0x80000000u) ? __uint_as_float(u & 0x7FFFFFFFu) : __uint_as_float(~u);
}

// ============================ WMMA GEMM ======================================
// OUT[M,dout] (+)= A[M,K] @ W[dout,K]^T  (+ bias) (relu)
// grid = (M/16, dout/16), block = 32 (one wave per 16x16 tile).
// f32 WMMA frag layout (ISA 7.12.2):
//   A 16x4 : lane<16 -> row=m0+lane, {K=k,k+1}; lane>=16 -> {K=k+2,k+3}
//   B 4x16 : lane%16 -> col, vgpr r/half select K (mirrors A pattern)
//   C 16x16: vgpr r, lanes0-15 -> M=r, lanes16-31 -> M=8+r, N=lane%16
template<bool ACC, bool BIAS, bool RELU>
__global__ void gemm_wmma(const float* __restrict__ A, const float* __restrict__ W,
                          const float* __restrict__ bias, float* __restrict__ OUT,
                          int K, int dout) {
  const int m0   = blockIdx.x * 16;
  const int n0   = blockIdx.y * 16;
  const int lane = threadIdx.x & 31;
  const int half = lane >> 4;
  const int lid  = lane & 15;
  const int col  = n0 + lid;
  float cr[8];
#if __has_builtin(__builtin_amdgcn_wmma_f32_16x16x4_f32)
  const float* ap = A + (size_t)(m0 + lid) * K + 2 * half;
  const float* bp = W + (size_t)col * K + 2 * half;
  v8f c = {0.f, 0.f, 0.f, 0.f, 0.f, 0.f, 0.f, 0.f};
  for (int k = 0; k < K; k += 4) {
    v2f a = *(const v2f*)(ap + k);
    v2f b = *(const v2f*)(bp + k);
    // 8 args: (neg_a, A, neg_b, B, c_mod, C, reuse_a, reuse_b)
    c = __builtin_amdgcn_wmma_f32_16x16x4_f32(false, a, false, b, (short)0, c,
                                              false, false);
  }
#pragma unroll
  for (int r = 0; r < 8; ++r) cr[r] = c[r];
#else
  // scalar fallback (keeps the build green if the builtin signature differs)
#pragma unroll
  for (int r = 0; r < 8; ++r) {
    const float* ar = A + (size_t)(m0 + r + 8 * half) * K;
    const float* wr = W + (size_t)col * K;
    float s = 0.f;
    for (int k = 0; k < K; ++k) s += ar[k] * wr[k];
    cr[r] = s;
  }
#endif
  const float bv = BIAS ? bias[col] : 0.0f;
#pragma unroll
  for (int r = 0; r < 8; ++r) {
    const size_t idx = (size_t)(m0 + r + 8 * half) * dout + col;
    float v = cr[r] + bv;
    if (ACC) v += OUT[idx];
    if (RELU) v = fmaxf(v, 0.0f);
    OUT[idx] = v;
  }
}

// ============================ small utilities ================================
__global__ void fill_f32(float* p, float v, int n) {
  int i = blockIdx.x * blockDim.x + threadIdx.x;
  if (i < n) p[i] = v;
}

__global__ void deg_accum(const int* __restrict__ dst, float* __restrict__ deg, int E) {
  int i = blockIdx.x * blockDim.x + threadIdx.x;
  if (i < E) atomicAdd(&deg[dst[i]], 1.0f);
}

__global__ void rsqrt_ip(float* p, int n) {
  int i = blockIdx.x * blockDim.x + threadIdx.x;
  if (i < n) p[i] = rsqrtf(p[i]);
}

// h0[n,128] = x[n,3] @ Winp[128,3]^T   (bias dropped: BN-invariant)
__global__ void input_linear(const float* __restrict__ x, const float* __restrict__ w,
                             float* __restrict__ out, int N) {
  int i = blockIdx.x * blockDim.x + threadIdx.x;
  if (i >= N * DD) return;
  int n = i >> 7, d = i & 127;
  const float* xr = x + n * 3;
  const float* wr = w + d * 3;
  out[i] = xr[0] * wr[0] + xr[1] * wr[1] + xr[2] * wr[2];
}

// ============================ BatchNorm ======================================
__global__ void bn_stats(const float* __restrict__ h, int n, int d,
                         float* __restrict__ sums, float* __restrict__ sqs) {
  int c = threadIdx.x;  // blockDim.x == d
  float s = 0.f, q = 0.f;
  for (int r = blockIdx.x; r < n; r += gridDim.x) {
    float v = h[(size_t)r * d + c];
    s += v;
    q += v * v;
  }
  atomicAdd(&sums[c], s);
  atomicAdd(&sqs[c], q);
}

__global__ void bn_finalize(const float* sums, const float* sqs, const float* g,
                            const float* b, float* scale, float* shift, int n, int d) {
  int c = threadIdx.x;
  if (c >= d) return;
  float inv_n = 1.0f / (float)n;
  float mu = sums[c] * inv_n;
  float var = sqs[c] * inv_n - mu * mu;
  float sc = g[c] * rsqrtf(var + 1e-5f);
  scale[c] = sc;
  shift[c] = b[c] - mu * sc;
}

// act: 0=none, 1=relu, 2=elu ; in-place
__global__ void bn_apply_act(float* h, const float* __restrict__ scale,
                             const float* __restrict__ shift, int n, int d, int act) {
  int i = blockIdx.x * blockDim.x + threadIdx.x;
  if (i >= n * d) return;
  int c = i % d;
  float v = h[i] * scale[c] + shift[c];
  if (act == 1) v = fmaxf(v, 0.0f);
  else if (act == 2) v = (v > 0.f) ? v : expm1f(v);
  h[i] = v;
}

// out = elu(bn(gat)) + elu(bn(gcn)) + id   (out may alias gat or id; elementwise)
__global__ void combine_dual_elu(const float* gat, const float* gcn, const float* idv,
                                 float* out, const float* __restrict__ s0,
                                 const float* __restrict__ t0,
                                 const float* __restrict__ s1,
                                 const float* __restrict__ t1, int n, int d) {
  int i = blockIdx.x * blockDim.x + threadIdx.x;
  if (i >= n * d) return;
  int c = i % d;
  float x = gat[i] * s0[c] + t0[c];
  x = (x > 0.f) ? x : expm1f(x);
  float y = gcn[i] * s1[c] + t1[c];
  y = (y > 0.f) ? y : expm1f(y);
  out[i] = x + y + idv[i];
}

// h = elu(bn(h)) + res   (in-place on h)
__global__ void bn_elu_add(float* h, const float* __restrict__ scale,
                           const float* __restrict__ shift,
                           const float* __restrict__ res, int n, int d) {
  int i = blockIdx.x * blockDim.x + threadIdx.x;
  if (i >= n * d) return;
  int c = i % d;
  float v = h[i] * scale[c] + shift[c];
  v = (v > 0.f) ? v : expm1f(v);
  h[i] = v + res[i];
}

// ============================ GAT ============================================
// per-(node,head) attention logits  a_s/a_d [N,8]
__global__ void gat_scores(const float* __restrict__ z, const float* __restrict__ asrc,
                           const float* __restrict__ adst, float* __restrict__ as_,
                           float* __restrict__ ad_, int N) {
  int i = blockIdx.x * blockDim.x + threadIdx.x;
  if (i >= N * HD) return;
  int n = i >> 3, hh = i & 7;
  const float* zp = z + (size_t)n * DD + hh * 16;
  const float* ap = asrc + hh * 16;
  const float* dp = adst + hh * 16;
  float s = 0.f, d = 0.f;
#pragma unroll
  for (int c = 0; c < 16; ++c) {
    float v = zp[c];
    s += v * ap[c];
    d += v * dp[c];
  }
  as_[i] = s;
  ad_[i] = d;
}

// segment max over dst (edges + self loops); mx stored enc_f-transformed
__global__ void gat_edge_max(const float* __restrict__ as_, const float* __restrict__ ad_,
                             const int* __restrict__ src, const int* __restrict__ dst,
                             unsigned* __restrict__ mx, int E, int N) {
  int i = blockIdx.x * blockDim.x + threadIdx.x;
  if (i >= (E + N) * HD) return;
  int e = i >> 3, hh = i & 7;
  int s = (e < E) ? src[e] : (e - E);
  int d = (e < E) ? dst[e] : (e - E);
  float x = as_[s * HD + hh] + ad_[d * HD + hh];
  x = (x > 0.f) ? x : 0.2f * x;  // leaky_relu(0.2)
  atomicMax(&mx[d * HD + hh], enc_f(x));
}

__global__ void gat_edge_sum(const float* __restrict__ as_, const float* __restrict__ ad_,
                             const int* __restrict__ src, const int* __restrict__ dst,
                             const unsigned* __restrict__ mx, float* __restrict__ sm,
                             int E, int N) {
  int i = blockIdx.x * blockDim.x + threadIdx.x;
  if (i >= (E + N) * HD) return;
  int e = i >> 3, hh = i & 7;
  int s = (e < E) ? src[e] : (e - E);
  int d = (e < E) ? dst[e] : (e - E);
  float x = as_[s * HD + hh] + ad_[d * HD + hh];
  x = (x > 0.f) ? x : 0.2f * x;
  atomicAdd(&sm[d * HD + hh], expf(x - dec_f(mx[d * HD + hh])));
}

// out[dst] += z[src] * alpha ; one wave per edge, 4 channels per lane
__global__ void gat_aggregate(const float* __restrict__ z, const float* __restrict__ as_,
                              const float* __restrict__ ad_, const unsigned* __restrict__ mx,
                              const float* __restrict__ sm, const int* __restrict__ src,
                              const int* __restrict__ dst, float* __restrict__ out,
                              int E, int N) {
  int eg = blockIdx.x * (blockDim.x >> 5) + (threadIdx.x >> 5);
  int lane = threadIdx.x & 31;
  if (eg >= E + N) return;
  int s = (eg < E) ? src[eg] : (eg - E);
  int d = (eg < E) ? dst[eg] : (eg - E);
  int hh = lane >> 2;  // 4 channels/lane, 16 channels/head
  float x = as_[s * HD + hh] + ad_[d * HD + hh];
  x = (x > 0.f) ? x : 0.2f * x;
  float alpha = expf(x - dec_f(mx[d * HD + hh])) / sm[d * HD + hh];
  int c0 = lane * 4;
  const float* zp = z + (size_t)s * DD + c0;
  float* op = out + (size_t)d * DD + c0;
  atomicAdd(op + 0, zp[0] * alpha);
  atomicAdd(op + 1, zp[1] * alpha);
  atomicAdd(op + 2, zp[2] * alpha);
  atomicAdd(op + 3, zp[3] * alpha);
}

// ============================ GCN / GraphConv aggregation ====================
// out[dst] += z[src] * dinv[src]*dinv[dst]  (edges + self loops), D = 128 or 64
__global__ void gcn_aggregate(const float* __restrict__ z, const float* __restrict__ dinv,
                              const int* __restrict__ src, const int* __restrict__ dst,
                              float* __restrict__ out, int E, int N, int D) {
  int eg = blockIdx.x * (blockDim.x >> 5) + (threadIdx.x >> 5);
  int lane = threadIdx.x & 31;
  if (eg >= E + N) return;
  int s = (eg < E) ? src[eg] : (eg - E);
  int d = (eg < E) ? dst[eg] : (eg - E);
  float coeff = dinv[s] * dinv[d];
  int cpl = D >> 5;
  const float* zp = z + (size_t)s * D + lane * cpl;
  float* op = out + (size_t)d * D + lane * cpl;
  for (int c = 0; c < cpl; ++c) atomicAdd(op + c, zp[c] * coeff);
}

// out[dst] += h[src]  (original edges only, D=128)
__global__ void plain_aggregate(const float* __restrict__ h, const int* __restrict__ src,
                                const int* __restrict__ dst, float* __restrict__ out, int E) {
  int eg = blockIdx.x * (blockDim.x >> 5) + (threadIdx.x >> 5);
  int lane = threadIdx.x & 31;
  if (eg >= E) return;
  int s = src[eg], d = dst[eg];
  int c0 = lane * 4;
  const float* hp = h + (size_t)s * DD + c0;
  float* op = out + (size_t)d * DD + c0;
  atomicAdd(op + 0, hp[0]);
  atomicAdd(op + 1, hp[1]);
  atomicAdd(op + 2, hp[2]);
  atomicAdd(op + 3, hp[3]);
}

// ============================ pooling readout ================================
__global__ void pool_accum(const float* __restrict__ h, const int* __restrict__ batch,
                           float* __restrict__ S, unsigned* __restrict__ MX,
                           float* __restrict__ counts, int N) {
  int n = blockIdx.x;
  int c = threadIdx.x;  // 64
  if (n >= N) return;
  int g = batch[n];
  float v = h[(size_t)n * 64 + c];
  atomicAdd(&S[g * 64 + c], v);
  atomicMax(&MX[g * 64 + c], enc_f(v));
  if (c == 0) atomicAdd(&counts[g], 1.0f);
}

__global__ void pool_build(const float* __restrict__ S, const unsigned* __restrict__ MX,
                           const float* __restrict__ counts, float* __restrict__ P, int G) {
  int i = blockIdx.x * blockDim.x + threadIdx.x;
  if (i >= G * 192) return;
  int g = i / 192, c = i % 192;
  float cnt = counts[g];
  float v;
  if (c < 64) v = S[g * 64 + c] / fmaxf(cnt, 1.0f);           // mean
  else if (c < 128) v = (cnt > 0.f) ? dec_f(MX[g * 64 + c - 64]) : 0.0f;  // max
  else v = S[g * 64 + c - 128];                                // sum
  P[i] = v;
}

// P2[g,:] *= sigmoid(P2[g,:] . aw + ab)   (one block of 128 per graph)
__global__ void att_scale(float* __restrict__ P2, const float* __restrict__ aw,
                          const float* __restrict__ ab, int G) {
  __shared__ float red[128];
  int g = blockIdx.x, t = threadIdx.x;
  float v = P2[(size_t)g * 128 + t];
  red[t] = v * aw[t];
  __syncthreads();
  for (int s = 64; s > 0; s >>= 1) {
    if (t < s) red[t] += red[t + s];
    __syncthreads();
  }
  float a = 1.0f / (1.0f + expf(-(red[0] + ab[0])));
  P2[(size_t)g * 128 + t] = v * a;
}

// final 16->6 linear + log_softmax
__global__ void head_logsoftmax(const float* __restrict__ T3, const float* __restrict__ w,
                                const float* __restrict__ b, float* __restrict__ out, int G) {
  int g = blockIdx.x * blockDim.x + threadIdx.x;
  if (g >= G) return;
  float lg[6];
  float mx = -1e30f;
#pragma unroll
  for (int j = 0; j < 6; ++j) {
    float s = b[j];
    const float* tr = T3 + g * 16;
    const float* wr = w + j * 16;
#pragma unroll
    for (int k = 0; k < 16; ++k) s += tr[k] * wr[k];
    lg[j] = s;
    mx = fmaxf(mx, s);
  }
  float se = 0.f;
#pragma unroll
  for (int j = 0; j < 6; ++j) se += expf(lg[j] - mx);
  float l = logf(se);
#pragma unroll
  for (int j = 0; j < 6; ++j) out[g * 6 + j] = lg[j] - mx - l;
}

// ============================ host orchestration =============================
enum {
  IN_X = 0, IN_EI = 1, IN_BATCH = 2,
  P_INP_W = 3, P_INP_B = 4, P_INBN_G = 5, P_INBN_B = 6,
  P_GAT1_W = 7, P_GAT1_AS = 8, P_GAT1_AD = 9, P_GAT1_B = 10,
  P_BN1_G = 11, P_BN1_B = 12,
  P_GCN1_W = 13, P_GCN1_B = 14, P_GCNBN1_G = 15, P_GCNBN1_B = 16,
  P_GAT2_W = 17, P_GAT2_AS = 18, P_GAT2_AD = 19, P_GAT2_B = 20,
  P_BN2_G = 21, P_BN2_B = 22,
  P_GCN2_W = 23, P_GCN2_B = 24, P_GCNBN2_G = 25, P_GCNBN2_B = 26,
  P_REL_W = 27, P_REL_B = 28, P_ROOT_W = 29,
  P_BN3_G = 30, P_BN3_B = 31,
  P_CONV4_W = 32, P_CONV4_B = 33, P_BN4_G = 34, P_BN4_B = 35,
  P_RES_W = 36, P_RES_B = 37,
  P_POOL_W = 38, P_POOL_B = 39,
  P_ATT_W = 40, P_ATT_B = 41,
  P_C1_W = 42, P_C1_B = 43, P_CBN1_G = 44, P_CBN1_B = 45,
  P_C2_W = 46, P_C2_B = 47, P_CBN2_G = 48, P_CBN2_B = 49,
  P_C3_W = 50, P_C3_B = 51, P_C4_W = 52, P_C4_B = 53
};

extern "C" void kernel_launch(void* const* d_in, const int* in_sizes, int n_in,
                              void* d_out, int out_size, void* d_ws, size_t ws_size,
                              hipStream_t stream) {
  const int N = in_sizes[IN_BATCH];       // 100000 (multiple of 16)
  const int E = in_sizes[IN_EI] / 2;      // 1600000
  const int G = out_size / 6;             // 512

  auto F = [&](int i) { return (const float*)d_in[i]; };
  const int* src = (const int*)d_in[IN_EI];
  const int* dst = src + E;
  const int* batch = (const int*)d_in[IN_BATCH];
  float* out = (float*)d_out;

  // ---- workspace carve (256B-aligned slices) ----
  float* ws = (float*)d_ws;
  size_t off = 0;
  auto carve = [&](size_t nf) {
    float* p = ws + off;
    off += (nf + 63) & ~(size_t)63;
    return p;
  };
  float* A    = carve((size_t)N * DD);   // h / id
  float* B    = carve((size_t)N * DD);   // gemm scratch / z
  float* C    = carve((size_t)N * DD);   // gat_out / h2 / out4
  float* Dh   = carve((size_t)N * DD);   // gcn_out / h3
  float* R    = carve((size_t)N * 64);   // residual [N,64]
  float* as_  = carve((size_t)N * HD);
  float* ad_  = carve((size_t)N * HD);
  unsigned* mx_ = (unsigned*)carve((size_t)N * HD);
  float* sm_  = carve((size_t)N * HD);
  float* dinv = carve((size_t)N);
  float* bnbuf = carve(1024);            // 2 slots x {sums,sqs,scale,shift}[128]
  float* counts = carve(G);
  float* Sg   = carve((size_t)G * 64);
  unsigned* MXg = (unsigned*)carve((size_t)G * 64);
  float* Pg   = carve((size_t)G * 192);
  float* P2   = carve((size_t)G * 128);
  float* T1   = carve((size_t)G * 64);
  float* T2   = carve((size_t)G * 32);
  float* T3   = carve((size_t)G * 16);
  (void)ws_size; (void)n_in;

  const int TB = 256;
  auto bn_prep = [&](const float* h, int n, int d, const float* g, const float* b,
                     float* slot) {
    float* sums = slot; float* sqs = slot + 128;
    float* scale = slot + 256; float* shift = slot + 384;
    hipMemsetAsync(sums, 0, 128 * sizeof(float), stream);
    hipMemsetAsync(sqs, 0, 128 * sizeof(float), stream);
    int blocks = n < 512 ? n : 512;
    bn_stats<<<blocks, d, 0, stream>>>(h, n, d, sums, sqs);
    bn_finalize<<<1, d, 0, stream>>>(sums, sqs, g, b, scale, shift, n, d);
  };
  float* sc0 = bnbuf + 256; float* sh0 = bnbuf + 384;
  float* sc1 = bnbuf + 512 + 256; float* sh1 = bnbuf + 512 + 384;

  const int egBlocks = cdiv_i(E + N, TB / 32);   // one wave per edge
  const int ehThreads = (E + N) * HD;

  // ---- degrees: deg = 1 + in-degree; dinv = rsqrt(deg) ----
  fill_f32<<<cdiv_i(N, TB), TB, 0, stream>>>(dinv, 1.0f, N);
  deg_accum<<<cdiv_i(E, TB), TB, 0, stream>>>(dst, dinv, E);
  rsqrt_ip<<<cdiv_i(N, TB), TB, 0, stream>>>(dinv, N);

  // ---- input layer: A = relu(in_bn(x @ Winp^T)) ----
  input_linear<<<cdiv_i(N * DD, TB), TB, 0, stream>>>(F(IN_X), F(P_INP_W), A, N);
  bn_prep(A, N, DD, F(P_INBN_G), F(P_INBN_B), bnbuf);
  bn_apply_act<<<cdiv_i(N * DD, TB), TB, 0, stream>>>(A, sc0, sh0, N, DD, 1);

  // ---- two GAT+GCN blocks ----
  for (int layer = 0; layer < 2; ++layer) {
    const float* gw  = F(layer ? P_GAT2_W : P_GAT1_W);
    const float* gas = F(layer ? P_GAT2_AS : P_GAT1_AS);
    const float* gad = F(layer ? P_GAT2_AD : P_GAT1_AD);
    const float* bng = F(layer ? P_BN2_G : P_BN1_G);
    const float* bnb = F(layer ? P_BN2_B : P_BN1_B);
    const float* cw  = F(layer ? P_GCN2_W : P_GCN1_W);
    const float* cbg = F(layer ? P_GCNBN2_G : P_GCNBN1_G);
    const float* cbb = F(layer ? P_GCNBN2_B : P_GCNBN1_B);

    // GAT: z = A @ gw^T
    gemm_wmma<false, false, false><<<dim3(N / 16, DD / 16), 32, 0, stream>>>(
        A, gw, nullptr, B, DD, DD);
    gat_scores<<<cdiv_i(N * HD, TB), TB, 0, stream>>>(B, gas, gad, as_, ad_, N);
    hipMemsetAsync(mx_, 0, (size_t)N * HD * 4, stream);   // enc lower bound
    hipMemsetAsync(sm_, 0, (size_t)N * HD * 4, stream);
    gat_edge_max<<<cdiv_i(ehThreads, TB), TB, 0, stream>>>(as_, ad_, src, dst, mx_, E, N);
    gat_edge_sum<<<cdiv_i(ehThreads, TB), TB, 0, stream>>>(as_, ad_, src, dst, mx_, sm_, E, N);
    hipMemsetAsync(C, 0, (size_t)N * DD * 4, stream);
    gat_aggregate<<<egBlocks, TB, 0, stream>>>(B, as_, ad_, mx_, sm_, src, dst, C, E, N);

    // GCN: z = A @ cw^T ; Dh = scatter(z * gcn_norm)
    gemm_wmma<false, false, false><<<dim3(N / 16, DD / 16), 32, 0, stream>>>(
        A, cw, nullptr, B, DD, DD);
    hipMemsetAsync(Dh, 0, (size_t)N * DD * 4, stream);
    gcn_aggregate<<<egBlocks, TB, 0, stream>>>(B, dinv, src, dst, Dh, E, N, DD);

    bn_prep(C, N, DD, bng, bnb, bnbuf);
    bn_prep(Dh, N, DD, cbg, cbb, bnbuf + 512);
    // layer0: A <- elu(bn(C)) + elu(bn(Dh)) + A      (A = h1)
    // layer1: C <- elu(bn(C)) + elu(bn(Dh)) + A      (C = h2, A stays id2)
    float* dest = layer ? C : A;
    combine_dual_elu<<<cdiv_i(N * DD, TB), TB, 0, stream>>>(C, Dh, A, dest, sc0, sh0,
                                                            sc1, sh1, N, DD);
  }

  // ---- graphconv (conv3): Dh = elu(bn3(agg@relW^T + h2@rootW^T)) ----
  hipMemsetAsync(B, 0, (size_t)N * DD * 4, stream);
  plain_aggregate<<<cdiv_i(E, TB / 32), TB, 0, stream>>>(C, src, dst, B, E);
  gemm_wmma<false, false, false><<<dim3(N / 16, DD / 16), 32, 0, stream>>>(
      B, F(P_REL_W), nullptr, Dh, DD, DD);
  gemm_wmma<true, false, false><<<dim3(N / 16, DD / 16), 32, 0, stream>>>(
      C, F(P_ROOT_W), nullptr, Dh, DD, DD);
  bn_prep(Dh, N, DD, F(P_BN3_G), F(P_BN3_B), bnbuf);
  bn_apply_act<<<cdiv_i(N * DD, TB), TB, 0, stream>>>(Dh, sc0, sh0, N, DD, 2);

  // ---- residual: R = id2 @ resW^T + resB  [N,64] ----
  gemm_wmma<false, true, false><<<dim3(N / 16, 64 / 16), 32, 0, stream>>>(
      A, F(P_RES_W), F(P_RES_B), R, DD, 64);

  // ---- conv4 (gcn 128->64): C[0:N*64] = elu(bn4(scatter(Dh@W^T))) + R ----
  gemm_wmma<false, false, false><<<dim3(N / 16, 64 / 16), 32, 0, stream>>>(
      Dh, F(P_CONV4_W), nullptr, B, DD, 64);
  hipMemsetAsync(C, 0, (size_t)N * 64 * 4, stream);
  gcn_aggregate<<<egBlocks, TB, 0, stream>>>(B, dinv, src, dst, C, E, N, 64);
  bn_prep(C, N, 64, F(P_BN4_G), F(P_BN4_B), bnbuf);
  bn_elu_add<<<cdiv_i(N * 64, TB), TB, 0, stream>>>(C, sc0, sh0, R, N, 64);

  // ---- multi-pool readout + attention gate ----
  hipMemsetAsync(counts, 0, G * 4, stream);
  hipMemsetAsync(Sg, 0, (size_t)G * 64 * 4, stream);
  hipMemsetAsync(MXg, 0, (size_t)G * 64 * 4, stream);
  pool_accum<<<N, 64, 0, stream>>>(C, batch, Sg, MXg, counts, N);
  pool_build<<<cdiv_i(G * 192, TB), TB, 0, stream>>>(Sg, MXg, counts, Pg, G);
  gemm_wmma<false, true, true><<<dim3(G / 16, DD / 16), 32, 0, stream>>>(
      Pg, F(P_POOL_W), F(P_POOL_B), P2, 192, DD);
  att_scale<<<G, 128, 0, stream>>>(P2, F(P_ATT_W), F(P_ATT_B), G);

  // ---- classifier head ----
  gemm_wmma<false, true, true><<<dim3(G / 16, 64 / 16), 32, 0, stream>>>(
      P2, F(P_C1_W), F(P_C1_B), T1, DD, 64);
  bn_prep(T1, G, 64, F(P_CBN1_G), F(P_CBN1_B), bnbuf);
  bn_apply_act<<<cdiv_i(G * 64, TB), TB, 0, stream>>>(T1, sc0, sh0, G, 64, 0);
  gemm_wmma<false, true, true><<<dim3(G / 16, 32 / 16), 32, 0, stream>>>(
      T1, F(P_C2_W), F(P_C2_B), T2, 64, 32);
  bn_prep(T2, G, 32, F(P_CBN2_G), F(P_CBN2_B), bnbuf);
  bn_apply_act<<<cdiv_i(G * 32, TB), TB, 0, stream>>>(T2, sc0, sh0, G, 32, 0);
  gemm_wmma<false, true, true><<<dim3(G / 16, 16 / 16), 32, 0, stream>>>(
      T2, F(P_C3_W), F(P_C3_B), T3, 32, 16);
  head_logsoftmax<<<cdiv_i(G, TB), TB, 0, stream>>>(T3, F(P_C4_W), F(P_C4_B), out, G);
}